// BaselineGCN_85899345950
// MI455X (gfx1250) — compile-verified
//
#include <hip/hip_runtime.h>
#include <hip/hip_bf16.h>

typedef __attribute__((ext_vector_type(2))) float v2f;
typedef __attribute__((ext_vector_type(8))) float v8f;

#define HID 64
#define GCOUNT 128
#define NCLASS 10
#define BN_EPS 1e-5f

// ---------------------------------------------------------------------------
// degree / normalization
// ---------------------------------------------------------------------------
__global__ void k_init_deg(float* deg, int n) {
    int i = blockIdx.x * blockDim.x + threadIdx.x;
    if (i < n) deg[i] = 1.0f;  // self-loop weight
}

__global__ void k_deg_edges(const int* __restrict__ dst, const float* __restrict__ w,
                            float* deg, int E) {
    int e = blockIdx.x * blockDim.x + threadIdx.x;
    if (e < E) atomicAdd(&deg[dst[e]], w[e]);
}

__global__ void k_dinv(const float* __restrict__ deg, float* dinv, int n) {
    int i = blockIdx.x * blockDim.x + threadIdx.x;
    if (i < n) {
        float d = deg[i];
        dinv[i] = d > 0.0f ? rsqrtf(d) : 0.0f;
    }
}

// ---------------------------------------------------------------------------
// fp32 WMMA GEMM: C[rows x 64] = A[rows x K] @ B[K x 64]
// block = 128 threads = 4 waves; each wave owns a 16x16 output tile
// (16 rows shared by the block, 16 distinct columns per wave).
// K in {64, 128}; rows divisible by 16.
// ---------------------------------------------------------------------------
__global__ void k_gemm_wmma(const float* __restrict__ A,
                            const float* __restrict__ B,
                            float* __restrict__ C,
                            int K) {
    __shared__ float Bs[128 * HID];  // max 32 KB (K<=128)
    const int tid = threadIdx.x;

    // cooperative stage of B into LDS
    for (int i = tid; i < K * HID; i += blockDim.x) Bs[i] = B[i];
    __syncthreads();

    const int wave = tid >> 5;
    const int lane = tid & 31;
    const int half = lane >> 4;      // 0: lanes 0-15, 1: lanes 16-31
    const int l16  = lane & 15;
    const int kOff = half * 2;       // ISA A/B fragment layout for 16x16x4 f32
    const int col  = wave * 16 + l16;
    const size_t rowBase = (size_t)blockIdx.x * 16;

    const float* Arow = A + (rowBase + l16) * (size_t)K;

    v8f acc = {};
    for (int k = 0; k < K; k += 4) {
        v2f a, b;
        a.x = Arow[k + kOff];
        a.y = Arow[k + kOff + 1];
        b.x = Bs[(k + kOff) * HID + col];
        b.y = Bs[(k + kOff + 1) * HID + col];
        acc = __builtin_amdgcn_wmma_f32_16x16x4_f32(
            /*neg_a=*/false, a, /*neg_b=*/false, b,
            /*c_mod=*/(short)0, acc, /*reuse_a=*/false, /*reuse_b=*/false);
    }

#pragma unroll
    for (int j = 0; j < 8; ++j) {
        size_t row = rowBase + (size_t)(j + half * 8);
        C[row * HID + col] = acc[j];
    }
}

// ---------------------------------------------------------------------------
// aggregation: out[i] = dinv[i]^2 * hw[i]   (self-loop term, also zero-inits)
// ---------------------------------------------------------------------------
__global__ void k_selfloop_init(const float* __restrict__ hw,
                                const float* __restrict__ dinv,
                                float* out, int total) {
    int idx = blockIdx.x * blockDim.x + threadIdx.x;
    if (idx < total) {
        int i = idx >> 6;  // /64
        float di = dinv[i];
        out[idx] = di * di * hw[idx];
    }
}

// per-edge scatter: out[dst] += dinv[src]*w*dinv[dst] * hw[src]
// one thread handles (edge, 2 channels)
__global__ void k_scatter(const float* __restrict__ hw,
                          const int* __restrict__ src,
                          const int* __restrict__ dst,
                          const float* __restrict__ w,
                          const float* __restrict__ dinv,
                          float* out, int E) {
    int tid = blockIdx.x * blockDim.x + threadIdx.x;
    int total = E * 32;
    if (tid >= total) return;
    int e = tid >> 5;
    int c = (tid & 31) * 2;
    int s = src[e], d = dst[e];
    float nrm = dinv[s] * w[e] * dinv[d];
    const float2 v = *(const float2*)(hw + (size_t)s * HID + c);
    atomicAdd(out + (size_t)d * HID + c,     nrm * v.x);
    atomicAdd(out + (size_t)d * HID + c + 1, nrm * v.y);
}

// fused bias + BatchNorm(eval) + ReLU, in place over [N, 64]
__global__ void k_bias_bn_relu(float* h,
                               const float* __restrict__ b,
                               const float* __restrict__ g,
                               const float* __restrict__ be,
                               const float* __restrict__ m,
                               const float* __restrict__ v,
                               int total) {
    int idx = blockIdx.x * blockDim.x + threadIdx.x;
    if (idx < total) {
        int c = idx & (HID - 1);
        float t = (h[idx] + b[c] - m[c]) * rsqrtf(v[c] + BN_EPS) * g[c] + be[c];
        h[idx] = t > 0.0f ? t : 0.0f;
    }
}

// ---------------------------------------------------------------------------
// pooling
// ---------------------------------------------------------------------------
__global__ void k_zero(float* p, int n) {
    int i = blockIdx.x * blockDim.x + threadIdx.x;
    if (i < n) p[i] = 0.0f;
}

__global__ void k_pool(const float* __restrict__ h, const int* __restrict__ batch,
                       float* pooled, int total) {
    int idx = blockIdx.x * blockDim.x + threadIdx.x;
    if (idx < total) {
        int i = idx >> 6;
        int c = idx & (HID - 1);
        atomicAdd(&pooled[batch[i] * HID + c], h[idx]);
    }
}

__global__ void k_cnt(const int* __restrict__ batch, float* cnt, int n) {
    int i = blockIdx.x * blockDim.x + threadIdx.x;
    if (i < n) atomicAdd(&cnt[batch[i]], 1.0f);
}

__global__ void k_div(float* pooled, const float* __restrict__ cnt, int total) {
    int idx = blockIdx.x * blockDim.x + threadIdx.x;
    if (idx < total) {
        float c = cnt[idx >> 6];  // /HID
        pooled[idx] /= fmaxf(c, 1.0f);
    }
}

// ---------------------------------------------------------------------------
// head: z = relu(BN(pooled @ Wf1 + bf1)); logits = z @ Wf2 + bf2; log_softmax
// one thread per graph (128 graphs)
// ---------------------------------------------------------------------------
__global__ void k_head(const float* __restrict__ pooled,
                       const float* __restrict__ Wf1, const float* __restrict__ bf1,
                       const float* __restrict__ gf, const float* __restrict__ bef,
                       const float* __restrict__ mf, const float* __restrict__ vf,
                       const float* __restrict__ Wf2, const float* __restrict__ bf2,
                       float* __restrict__ out) {
    int g = blockIdx.x * blockDim.x + threadIdx.x;
    if (g >= GCOUNT) return;

    float z[32];
#pragma unroll 4
    for (int j = 0; j < 32; ++j) {
        float s = bf1[j];
        for (int c = 0; c < HID; ++c) s += pooled[g * HID + c] * Wf1[c * 32 + j];
        s = (s - mf[j]) * rsqrtf(vf[j] + BN_EPS) * gf[j] + bef[j];
        z[j] = s > 0.0f ? s : 0.0f;
    }

    float logit[NCLASS];
    float mx = -3.4e38f;
#pragma unroll
    for (int k = 0; k < NCLASS; ++k) {
        float s = bf2[k];
        for (int j = 0; j < 32; ++j) s += z[j] * Wf2[j * NCLASS + k];
        logit[k] = s;
        mx = fmaxf(mx, s);
    }
    float sum = 0.0f;
#pragma unroll
    for (int k = 0; k < NCLASS; ++k) sum += expf(logit[k] - mx);
    float lse = logf(sum);
#pragma unroll
    for (int k = 0; k < NCLASS; ++k) out[g * NCLASS + k] = logit[k] - mx - lse;
}

// ---------------------------------------------------------------------------
extern "C" void kernel_launch(void* const* d_in, const int* in_sizes, int n_in,
                              void* d_out, int out_size, void* d_ws, size_t ws_size,
                              hipStream_t stream) {
    const float* x     = (const float*)d_in[0];   // [N, 128]
    const int*   ei    = (const int*)  d_in[1];   // [2, E]
    const int*   batch = (const int*)  d_in[2];   // [N]
    const float* ea    = (const float*)d_in[3];   // [E]
    // d_in[4] = pos, unused by the reference math
    const float* W0  = (const float*)d_in[5];
    const float* b0  = (const float*)d_in[6];
    const float* g0  = (const float*)d_in[7];
    const float* be0 = (const float*)d_in[8];
    const float* m0  = (const float*)d_in[9];
    const float* v0  = (const float*)d_in[10];
    const float* W1  = (const float*)d_in[11];
    const float* b1  = (const float*)d_in[12];
    const float* g1  = (const float*)d_in[13];
    const float* be1 = (const float*)d_in[14];
    const float* m1  = (const float*)d_in[15];
    const float* v1  = (const float*)d_in[16];
    const float* Wf1 = (const float*)d_in[17];
    const float* bf1 = (const float*)d_in[18];
    const float* gf  = (const float*)d_in[19];
    const float* bef = (const float*)d_in[20];
    const float* mf  = (const float*)d_in[21];
    const float* vf  = (const float*)d_in[22];
    const float* Wf2 = (const float*)d_in[23];
    const float* bf2 = (const float*)d_in[24];

    const int N = in_sizes[2];        // 100000
    const int E = in_sizes[3];        // 1000000
    const int K0 = in_sizes[0] / N;   // 128

    const int* src = ei;
    const int* dst = ei + E;

    float* ws     = (float*)d_ws;
    float* deg    = ws;                           // [N]
    float* dinv   = deg + N;                      // [N]
    float* bufA   = dinv + N;                     // [N, 64]
    float* bufB   = bufA + (size_t)N * HID;       // [N, 64]
    float* pooled = bufB + (size_t)N * HID;       // [G, 64]
    float* cnt    = pooled + GCOUNT * HID;        // [G]

    const int T = 256;
    const int nB   = (N + T - 1) / T;
    const int eB   = (E + T - 1) / T;
    const int nhB  = ((N * HID) + T - 1) / T;     // N*64 = 6.4M -> fits int
    const int scB  = ((E * 32) + T - 1) / T;      // E*32 = 32M -> fits int

    // --- normalization coefficients ---
    k_init_deg<<<nB, T, 0, stream>>>(deg, N);
    k_deg_edges<<<eB, T, 0, stream>>>(dst, ea, deg, E);
    k_dinv<<<nB, T, 0, stream>>>(deg, dinv, N);

    // --- layer 0: x @ W0 -> aggregate -> bias+BN+ReLU ---
    k_gemm_wmma<<<N / 16, 128, 0, stream>>>(x, W0, bufA, K0);
    k_selfloop_init<<<nhB, T, 0, stream>>>(bufA, dinv, bufB, N * HID);
    k_scatter<<<scB, T, 0, stream>>>(bufA, src, dst, ea, dinv, bufB, E);
    k_bias_bn_relu<<<nhB, T, 0, stream>>>(bufB, b0, g0, be0, m0, v0, N * HID);

    // --- layer 1: h @ W1 -> aggregate -> bias+BN+ReLU ---
    k_gemm_wmma<<<N / 16, 128, 0, stream>>>(bufB, W1, bufA, HID);
    k_selfloop_init<<<nhB, T, 0, stream>>>(bufA, dinv, bufB, N * HID);
    k_scatter<<<scB, T, 0, stream>>>(bufA, src, dst, ea, dinv, bufB, E);
    k_bias_bn_relu<<<nhB, T, 0, stream>>>(bufB, b1, g1, be1, m1, v1, N * HID);

    // --- global mean pool ---
    k_zero<<<(GCOUNT * HID + GCOUNT + T - 1) / T, T, 0, stream>>>(pooled, GCOUNT * HID + GCOUNT);
    k_pool<<<nhB, T, 0, stream>>>(bufB, batch, pooled, N * HID);
    k_cnt<<<nB, T, 0, stream>>>(batch, cnt, N);
    k_div<<<(GCOUNT * HID + T - 1) / T, T, 0, stream>>>(pooled, cnt, GCOUNT * HID);

    // --- MLP head + log_softmax ---
    k_head<<<1, GCOUNT, 0, stream>>>(pooled, Wf1, bf1, gf, bef, mf, vf, Wf2, bf2,
                                     (float*)d_out);
}